// CompactBilinearPooling_81870666597053
// MI455X (gfx1250) — compile-verified
//
#include <hip/hip_runtime.h>
#include <hip/hip_bf16.h>

typedef __attribute__((ext_vector_type(2))) float v2f;
typedef __attribute__((ext_vector_type(8))) float v8f;

#define B_   16
#define C_   512
#define HW_  196
#define OUT_ 8192

// ---------------------------------------------------------------------------
// Stage 0: recover the sparse count-sketch (one nonzero +-1 per row).
// 1024 blocks: block = (sketch index << 9) | row. Exactly one lane finds the
// nonzero, so it stores without any race.
// ---------------------------------------------------------------------------
__global__ void sketch_extract_kernel(const float* __restrict__ sk1,
                                      const float* __restrict__ sk2,
                                      int* __restrict__ h1, float* __restrict__ s1,
                                      int* __restrict__ h2, float* __restrict__ s2) {
    const int blk   = blockIdx.x;
    const int which = blk >> 9;
    const int row   = blk & (C_ - 1);
    const float* S  = which ? sk2 : sk1;
    int*   h        = which ? h2  : h1;
    float* s        = which ? s2  : s1;
    const float* rowp = S + (size_t)row * OUT_;
    for (int col = threadIdx.x; col < OUT_; col += blockDim.x) {
        float v = rowp[col];
        if (v != 0.0f) { h[row] = col; s[row] = v; }
    }
}

// ---------------------------------------------------------------------------
// Stage 1: G[b] = X[b] * X[b]^T via V_WMMA_F32_16X16X4_F32.
// One wave computes one 16x16 tile of G; K loop over 196 spatial positions in
// steps of 4 (49 WMMAs per tile).
//
// fp32 A (16x4, MxK) VGPR layout: lane holds row M = lane&15;
//   lanes 0-15: {K=0,K=1}, lanes 16-31: {K=2,K=3}  -> kOff = (lane>>4)*2.
// fp32 B (4x16, KxN) is X^T, so B[k][n] = X[j0+n][k]: same load pattern with
// the column-tile base row.
// fp32 C/D (16x16): VGPR v, lane l -> M = v + 8*(l>>4), N = l&15.
// ---------------------------------------------------------------------------
__global__ void gram_wmma_kernel(const float* __restrict__ x, float* __restrict__ G) {
    const int lane         = threadIdx.x & 31;
    const int waveInBlock  = threadIdx.x >> 5;
    const int wavesPerBlk  = blockDim.x >> 5;
    const int tile         = blockIdx.x * wavesPerBlk + waveInBlock;

    const int b  = tile >> 10;          // 1024 tiles per batch (32x32 grid of 16x16)
    const int t  = tile & 1023;
    const int i0 = (t >> 5) << 4;
    const int j0 = (t & 31) << 4;

    const float* Xb  = x + (size_t)b * C_ * HW_;
    const int rA     = i0 + (lane & 15);
    const int rB     = j0 + (lane & 15);
    const int kOff   = (lane >> 4) << 1;
    const float* pA  = Xb + (size_t)rA * HW_ + kOff;
    const float* pB  = Xb + (size_t)rB * HW_ + kOff;

    v8f acc = {};
#pragma unroll 7
    for (int k = 0; k < HW_; k += 4) {
        v2f a  = *(const v2f*)(pA + k);
        v2f bm = *(const v2f*)(pB + k);
        acc = __builtin_amdgcn_wmma_f32_16x16x4_f32(
            /*neg_a=*/false, a, /*neg_b=*/false, bm,
            /*c_mod=*/(short)0, acc, /*reuse_a=*/false, /*reuse_b=*/false);
    }

    float* Gt       = G + ((size_t)b * C_ + i0) * C_ + j0;
    const int n     = lane & 15;
    const int mBase = (lane >> 4) << 3;
#pragma unroll
    for (int v = 0; v < 8; ++v)
        Gt[(size_t)(mBase + v) * C_ + n] = acc[v];
}

// ---------------------------------------------------------------------------
// Stage 2+3 (fused): per batch, scatter G[b,c1,c2]*s1*s2 into an 8192-bin LDS
// accumulator at (h1[c1]+h2[c2]) & 8191 (count-sketch circular conv + spatial
// sum-pool), then signed-sqrt, L2-normalize, write out. One 1024-thread block
// per batch; LDS atomics (ds_add_f32), zero global atomics.
// ---------------------------------------------------------------------------
__global__ void cbp_accum_finalize_kernel(const float* __restrict__ G,
                                          const int* __restrict__ h1, const float* __restrict__ s1,
                                          const int* __restrict__ h2, const float* __restrict__ s2,
                                          float* __restrict__ out) {
    __shared__ float acc[OUT_];
    __shared__ float red[1024];

    const int b    = blockIdx.x;
    const int tid  = threadIdx.x;
    const int nthr = blockDim.x;

    for (int o = tid; o < OUT_; o += nthr) acc[o] = 0.0f;
    __syncthreads();

    const float* Gb = G + (size_t)b * C_ * C_;
    for (int p = tid; p < C_ * C_; p += nthr) {
        const int c1  = p >> 9;
        const int c2  = p & (C_ - 1);
        const float v = Gb[p] * s1[c1] * s2[c2];
        const int o   = (h1[c1] + h2[c2]) & (OUT_ - 1);
        atomicAdd(&acc[o], v);
    }
    __syncthreads();

    // signed sqrt + partial sum of squares
    float ss = 0.0f;
    for (int o = tid; o < OUT_; o += nthr) {
        float v  = acc[o];
        float sg = (v > 0.0f) ? 1.0f : ((v < 0.0f) ? -1.0f : 0.0f);
        float t  = sg * sqrtf(fabsf(v) + 1e-5f);
        acc[o]   = t;
        ss      += t * t;
    }
    red[tid] = ss;
    __syncthreads();
    for (int s = nthr >> 1; s > 0; s >>= 1) {
        if (tid < s) red[tid] += red[tid + s];
        __syncthreads();
    }

    const float norm = sqrtf(red[0]);
    const float inv  = 1.0f / fmaxf(norm, 1e-12f);
    for (int o = tid; o < OUT_; o += nthr)
        out[(size_t)b * OUT_ + o] = acc[o] * inv;
}

// ---------------------------------------------------------------------------
extern "C" void kernel_launch(void* const* d_in, const int* in_sizes, int n_in,
                              void* d_out, int out_size, void* d_ws, size_t ws_size,
                              hipStream_t stream) {
    const float* x   = (const float*)d_in[0];   // [16,512,14,14]
    const float* sk1 = (const float*)d_in[1];   // [512,8192]
    const float* sk2 = (const float*)d_in[2];   // [512,8192]
    float* out       = (float*)d_out;           // [16,8192]

    // Workspace layout: h1,h2 (512 i32 each), s1,s2 (512 f32 each), then G.
    char*  ws = (char*)d_ws;
    int*   h1 = (int*)ws;
    int*   h2 = h1 + C_;
    float* s1 = (float*)(h2 + C_);
    float* s2 = s1 + C_;
    float* G  = s2 + C_;                        // 16*512*512 f32 = 16 MiB

    // Stage 0: sparse sketch recovery. 2 sketches x 512 rows.
    sketch_extract_kernel<<<2 * C_, 256, 0, stream>>>(sk1, sk2, h1, s1, h2, s2);

    // Stage 1: WMMA Gram matrices. 16 batches * 1024 tiles, 4 waves/block.
    gram_wmma_kernel<<<(B_ * 1024) / 4, 128, 0, stream>>>(x, G);

    // Stage 2+3: sketch-space accumulation + signed sqrt + L2 normalize.
    cbp_accum_finalize_kernel<<<B_, 1024, 0, stream>>>(G, h1, s1, h2, s2, out);
}